// GCNConv_88244398064424
// MI455X (gfx1250) — compile-verified
//
#include <hip/hip_runtime.h>

typedef float v2f __attribute__((ext_vector_type(2)));
typedef float v8f __attribute__((ext_vector_type(8)));

#define N_NODES 50000
#define N_EDGES 1600000
#define IN_CH   128
#define EDGE_DIM 32
#define K_DIM   160          // IN_CH + EDGE_DIM
#define OUT_CH  128
#define LDS_STRIDE 164       // padded A-tile stride: conflict-free & even (8B-aligned f2)

// ---------------------------------------------------------------------------
// Phase 1: zero the agg scratch buffer (50000 x 160 f32 = 32 MB in d_ws)
// ---------------------------------------------------------------------------
__global__ void zero_f4(float4* __restrict__ p, int n4) {
    int i = blockIdx.x * blockDim.x + threadIdx.x;
    if (i < n4) p[i] = make_float4(0.f, 0.f, 0.f, 0.f);
}

// ---------------------------------------------------------------------------
// Phase 2: one wave32 per edge. Lane l handles x-channels {l, l+32, l+64, l+96}
// and edge channel l. Gathers of x[col] (25.6MB) and the atomic targets (32MB)
// are L2-resident; all loads/atomics are lane-coalesced. Hardware f32 atomics
// via unsafeAtomicAdd -> global_atomic_add_f32.
// ---------------------------------------------------------------------------
__global__ __launch_bounds__(256) void edge_scatter(
    const float* __restrict__ x, const int* __restrict__ row,
    const int* __restrict__ col, const float* __restrict__ norm,
    const float* __restrict__ ea, float* __restrict__ agg) {
    const int wave = (blockIdx.x * 256 + threadIdx.x) >> 5;
    const int lane = threadIdx.x & 31;
    if (wave >= N_EDGES) return;
    const int   r  = row[wave];
    const int   c  = col[wave];
    const float nv = norm[wave];
    const float* xr = x + (size_t)c * IN_CH;
    float*       ar = agg + (size_t)r * K_DIM;
#pragma unroll
    for (int j = 0; j < 4; ++j) {
        unsafeAtomicAdd(&ar[lane + 32 * j], nv * xr[lane + 32 * j]);
    }
    unsafeAtomicAdd(&ar[IN_CH + lane], nv * ea[(size_t)wave * EDGE_DIM + lane]);
}

// ---------------------------------------------------------------------------
// Phase 3: out[50000x128] = agg[50000x160] @ W[160x128] + b, exact f32 via
// V_WMMA_F32_16X16X4_F32. One block = one 16-row tile; 8 waves cover the 8
// column tiles; A tile (16x160) staged in LDS with padded stride.
//
// A 16x4 frag layout: lanes 0-15 -> M=lane, VGPR{0,1}=K{0,1};
//                     lanes 16-31 -> M=lane-16, VGPR{0,1}=K{2,3}.
// B 4x16 frag mirrored. C/D: VGPR j, lanes 0-15 -> M=j, lanes 16-31 -> M=j+8.
// ---------------------------------------------------------------------------
__global__ __launch_bounds__(256) void gemm_wmma(
    const float* __restrict__ agg, const float* __restrict__ W,
    const float* __restrict__ bias, float* __restrict__ out) {
    __shared__ float lds_a[16 * LDS_STRIDE];

    const int tid   = threadIdx.x;
    const int rtile = blockIdx.x;

    // Cooperative coalesced stage of the 16x160 A tile into padded LDS.
    const float* asrc = agg + (size_t)rtile * 16 * K_DIM;
#pragma unroll
    for (int i = 0; i < (16 * K_DIM) / 256; ++i) {
        int idx = tid + i * 256;
        int m   = idx / K_DIM;
        int k   = idx - m * K_DIM;
        lds_a[m * LDS_STRIDE + k] = asrc[idx];
    }
    __syncthreads();

    const int ctile = tid >> 5;        // 0..7 -> output column tile
    const int lane  = tid & 31;
    const int hi    = lane >> 4;       // lane half selects K pair
    const int l15   = lane & 15;
    const int ncol  = ctile * 16 + l15;

    v8f c = {};
#pragma unroll 4
    for (int k = 0; k < K_DIM; k += 4) {
        // A fragment: two consecutive K values for row l15 (8B-aligned in LDS)
        v2f a = *(const v2f*)&lds_a[l15 * LDS_STRIDE + k + 2 * hi];
        // B fragment: W rows (k+2*hi), (k+2*hi+1), column ncol (L2-resident)
        v2f b;
        b.x = W[(size_t)(k + 2 * hi)     * OUT_CH + ncol];
        b.y = W[(size_t)(k + 2 * hi + 1) * OUT_CH + ncol];
        c = __builtin_amdgcn_wmma_f32_16x16x4_f32(
                /*neg_a=*/false, a, /*neg_b=*/false, b,
                /*c_mod=*/(short)0, c, /*reuse_a=*/false, /*reuse_b=*/false);
    }

    const float bv    = bias[ncol];
    const int   mbase = rtile * 16 + hi * 8;
#pragma unroll
    for (int j = 0; j < 8; ++j) {
        out[(size_t)(mbase + j) * OUT_CH + ncol] = c[j] + bv;
    }
}

// ---------------------------------------------------------------------------
extern "C" void kernel_launch(void* const* d_in, const int* in_sizes, int n_in,
                              void* d_out, int out_size, void* d_ws, size_t ws_size,
                              hipStream_t stream) {
    const float* x    = (const float*)d_in[0];
    const int*   row  = (const int*)  d_in[1];
    const int*   col  = (const int*)  d_in[2];
    const float* norm = (const float*)d_in[3];
    const float* ea   = (const float*)d_in[4];
    const float* W    = (const float*)d_in[5];
    const float* bias = (const float*)d_in[6];
    float*       out  = (float*)d_out;
    float*       agg  = (float*)d_ws;   // 50000*160 f32 = 32 MB scratch

    const int n4 = (N_NODES * K_DIM) / 4;                     // 2,000,000 float4
    zero_f4<<<(n4 + 255) / 256, 256, 0, stream>>>((float4*)agg, n4);

    edge_scatter<<<N_EDGES / 8, 256, 0, stream>>>(x, row, col, norm, ea, agg);

    gemm_wmma<<<N_NODES / 16, 256, 0, stream>>>(agg, W, bias, out);
}